// GraphAttention_23905787969917
// MI455X (gfx1250) — compile-verified
//
#include <hip/hip_runtime.h>

#define GAT_ALPHA 0.2f

typedef __attribute__((ext_vector_type(2))) float v2f;
typedef __attribute__((ext_vector_type(8))) float v8f;

// ---- order-preserving float<->int map so float max can use integer atomicMax
__device__ __forceinline__ int f32_ord(float f) {
    int i = __float_as_int(f);
    return i >= 0 ? i : (i ^ 0x7FFFFFFF);
}
__device__ __forceinline__ float f32_unord(int i) {
    return __int_as_float(i >= 0 ? i : (i ^ 0x7FFFFFFF));
}
__device__ __forceinline__ float leaky(float x) { return x > 0.f ? x : GAT_ALPHA * x; }

// ---------------------------------------------------------------------------
// Init: out = 0, denom = 0, emax = -inf (encoded as INT_MIN in ordered space)
// ---------------------------------------------------------------------------
__global__ void gat_init(float* __restrict__ out, int* __restrict__ emax_ord,
                         float* __restrict__ denom, int N) {
    const int total = N * 128;
    for (int i = blockIdx.x * blockDim.x + threadIdx.x; i < total;
         i += gridDim.x * blockDim.x) {
        out[i] = 0.f;
        if (i < N) {
            emax_ord[i] = (int)0x80000000;  // minimum in ordered-int space
            denom[i]    = 0.f;
        }
    }
}

// ---------------------------------------------------------------------------
// GEMM: ft[N,128] = inputs[N,128] @ fc_w[128,128] via V_WMMA_F32_16X16X4_F32.
//
// fc_w staged in LDS in K-pair-interleaved layout:
//   Bs[(k>>1)*256 + n*2 + (k&1)] = B[k][n]
// so each WMMA B fragment (B[k][n], B[k+1][n]) is one aligned ds_load_b64
// straight into a contiguous VGPR pair (no v_mov shuffles).
//
// One wave computes a full 16x128 output strip: two K sweeps of 4 accumulators each,
// A fragment (global_load_b64) reused by 4 WMMAs per k-step.
// ---------------------------------------------------------------------------
__global__ __launch_bounds__(256) void gat_gemm(const float* __restrict__ A,
                                                const float* __restrict__ B,
                                                float* __restrict__ ft, int N) {
    __shared__ float Bs[128 * 128];  // 64 KB, interleaved layout
    // Cooperative staging: thread handles (k-pair kp, col n); two coalesced
    // global loads 512B apart, one 8B-aligned ds_store_b64.
    for (int idx = threadIdx.x; idx < 64 * 128; idx += 256) {
        const int kp = idx >> 7;
        const int n  = idx & 127;
        float2 v;
        v.x = B[(2 * kp) * 128 + n];
        v.y = B[(2 * kp + 1) * 128 + n];
        *(float2*)&Bs[kp * 256 + n * 2] = v;
    }
    __syncthreads();

    const int wave = threadIdx.x >> 5;
    const int lane = threadIdx.x & 31;
    const int row0 = (blockIdx.x * 8 + wave) * 16;
    if (row0 >= N) return;  // wave-uniform: EXEC stays all-ones for WMMA

    const int m  = lane & 15;          // M (A) / N (B,C) index within tile
    const int kh = (lane >> 4) << 1;   // K sub-offset: lanes 0-15 -> 0, 16-31 -> 2
    const float* __restrict__ arow = A + (size_t)(row0 + m) * 128;

    v8f acc0 = {}, acc1 = {}, acc2 = {}, acc3 = {};
#pragma unroll
    for (int k0 = 0; k0 < 128; k0 += 4) {
        // A 16x4 f32 frag: VGPR0 = K=(kh), VGPR1 = K=(kh+1); contiguous b64 load
        const v2f a = *(const v2f*)&arow[k0 + kh];
        // B 4x16 frags: one ds_load_b64 each from interleaved LDS
        const float* bk = &Bs[((k0 + kh) >> 1) * 256 + m * 2];
        const v2f b0 = *(const v2f*)&bk[0 * 32];
        const v2f b1 = *(const v2f*)&bk[16 * 2];
        const v2f b2 = *(const v2f*)&bk[32 * 2];
        const v2f b3 = *(const v2f*)&bk[48 * 2];
        acc0 = __builtin_amdgcn_wmma_f32_16x16x4_f32(false, a, false, b0,
                                                     (short)0, acc0, false, false);
        acc1 = __builtin_amdgcn_wmma_f32_16x16x4_f32(false, a, false, b1,
                                                     (short)0, acc1, false, false);
        acc2 = __builtin_amdgcn_wmma_f32_16x16x4_f32(false, a, false, b2,
                                                     (short)0, acc2, false, false);
        acc3 = __builtin_amdgcn_wmma_f32_16x16x4_f32(false, a, false, b3,
                                                     (short)0, acc3, false, false);
    }

    // C layout: VGPR v -> M = v + 8*(lane/16), N = tile_n0 + (lane%16)
    const int mhi = (lane >> 4) << 3;
    float* crow = ft + (size_t)row0 * 128 + m;
#pragma unroll
    for (int v = 0; v < 8; ++v) {
        float* r = crow + (size_t)(v + mhi) * 128;
        r[0]  = acc0[v];
        r[16] = acc1[v];
        r[32] = acc2[v];
        r[48] = acc3[v];
    }
    // second half of the 128 columns: tiles 4..7 share the same Bs sweep?
    // No -- 4 tiles cover 64 cols; run the second sweep for cols 64..127.
    v8f acc4 = {}, acc5 = {}, acc6 = {}, acc7 = {};
#pragma unroll
    for (int k0 = 0; k0 < 128; k0 += 4) {
        const v2f a = *(const v2f*)&arow[k0 + kh];
        const float* bk = &Bs[((k0 + kh) >> 1) * 256 + (64 + m) * 2];
        const v2f b0 = *(const v2f*)&bk[0 * 32];
        const v2f b1 = *(const v2f*)&bk[16 * 2];
        const v2f b2 = *(const v2f*)&bk[32 * 2];
        const v2f b3 = *(const v2f*)&bk[48 * 2];
        acc4 = __builtin_amdgcn_wmma_f32_16x16x4_f32(false, a, false, b0,
                                                     (short)0, acc4, false, false);
        acc5 = __builtin_amdgcn_wmma_f32_16x16x4_f32(false, a, false, b1,
                                                     (short)0, acc5, false, false);
        acc6 = __builtin_amdgcn_wmma_f32_16x16x4_f32(false, a, false, b2,
                                                     (short)0, acc6, false, false);
        acc7 = __builtin_amdgcn_wmma_f32_16x16x4_f32(false, a, false, b3,
                                                     (short)0, acc7, false, false);
    }
#pragma unroll
    for (int v = 0; v < 8; ++v) {
        float* r = crow + (size_t)(v + mhi) * 128 + 64;
        r[0]  = acc4[v];
        r[16] = acc5[v];
        r[32] = acc6[v];
        r[48] = acc7[v];
    }
}

// ---------------------------------------------------------------------------
// Per-node head-0 attention scores: a1 = ft[n,0:32].attn_l, a2 = ..attn_r
// ---------------------------------------------------------------------------
__global__ void gat_scores(const float* __restrict__ ft,
                           const float* __restrict__ attn_l,
                           const float* __restrict__ attn_r,
                           float* __restrict__ a1, float* __restrict__ a2, int N) {
    const int n = blockIdx.x * blockDim.x + threadIdx.x;
    if (n >= N) return;
    const float4* f  = (const float4*)(ft + (size_t)n * 128);
    const float4* al = (const float4*)attn_l;
    const float4* ar = (const float4*)attn_r;
    float s1 = 0.f, s2 = 0.f;
#pragma unroll
    for (int q = 0; q < 8; ++q) {
        const float4 v = f[q], l = al[q], r = ar[q];
        s1 += v.x * l.x + v.y * l.y + v.z * l.z + v.w * l.w;
        s2 += v.x * r.x + v.y * r.y + v.z * r.z + v.w * r.w;
    }
    a1[n] = s1;
    a2[n] = s2;
}

// ---------------------------------------------------------------------------
// Edge pass 1: segment max of leaky-relu logits into emax_ord[dst]
// ---------------------------------------------------------------------------
__global__ void gat_edge_max(const int* __restrict__ src, const int* __restrict__ dst,
                             const float* __restrict__ a1, const float* __restrict__ a2,
                             int* __restrict__ emax_ord, int E) {
    const int i = blockIdx.x * blockDim.x + threadIdx.x;
    if (i >= E) return;
    const int d = dst[i];
    const float e = leaky(a1[src[i]] + a2[d]);
    atomicMax(&emax_ord[d], f32_ord(e));
}

// ---------------------------------------------------------------------------
// Edge pass 2: denom[dst] += exp(e - emax[dst])
// ---------------------------------------------------------------------------
__global__ void gat_edge_denom(const int* __restrict__ src, const int* __restrict__ dst,
                               const float* __restrict__ a1, const float* __restrict__ a2,
                               const int* __restrict__ emax_ord,
                               float* __restrict__ denom, int E) {
    const int i = blockIdx.x * blockDim.x + threadIdx.x;
    if (i >= E) return;
    const int d = dst[i];
    const float e = leaky(a1[src[i]] + a2[d]);
    const float ex = __expf(e - f32_unord(emax_ord[d]));
    atomicAdd(&denom[d], ex);
}

// ---------------------------------------------------------------------------
// Edge pass 3: out[dst] += ft[src] * softmax_weight. One wave per edge,
// one float4 (4 dims) per lane -> 128 dims per wave. The edge index is
// forced into an SGPR (readfirstlane) so all the per-edge scalars become
// wave-uniform scalar loads instead of 32x-redundant vector loads.
// ---------------------------------------------------------------------------
__global__ __launch_bounds__(256) void gat_aggregate(
        const int* __restrict__ src, const int* __restrict__ dst,
        const float* __restrict__ a1, const float* __restrict__ a2,
        const int* __restrict__ emax_ord, const float* __restrict__ denom,
        const float* __restrict__ ft, float* __restrict__ out, int E) {
    const int i = __builtin_amdgcn_readfirstlane(blockIdx.x * 8 + (threadIdx.x >> 5));
    if (i >= E) return;  // wave-uniform guard
    const int lane = threadIdx.x & 31;

    const int s = src[i], d = dst[i];
    const float e = leaky(a1[s] + a2[d]);
    const float w = __expf(e - f32_unord(emax_ord[d])) / denom[d];

    const float4 v = ((const float4*)(ft + (size_t)s * 128))[lane];
    float* o = out + (size_t)d * 128 + lane * 4;
    atomicAdd(o + 0, v.x * w);
    atomicAdd(o + 1, v.y * w);
    atomicAdd(o + 2, v.z * w);
    atomicAdd(o + 3, v.w * w);
}

// ---------------------------------------------------------------------------
extern "C" void kernel_launch(void* const* d_in, const int* in_sizes, int n_in,
                              void* d_out, int out_size, void* d_ws, size_t ws_size,
                              hipStream_t stream) {
    const float* inputs = (const float*)d_in[0];
    const int*   src    = (const int*)d_in[1];
    const int*   dst    = (const int*)d_in[2];
    const float* fc_w   = (const float*)d_in[3];
    const float* attn_l = (const float*)d_in[4];
    const float* attn_r = (const float*)d_in[5];
    float* out = (float*)d_out;

    const int N = in_sizes[0] / 128;
    const int E = in_sizes[1];

    // workspace carve-up (~53 MB)
    char* p = (char*)d_ws;
    float* ft       = (float*)p;  p += (size_t)N * 128 * sizeof(float);
    float* a1       = (float*)p;  p += (size_t)N * sizeof(float);
    float* a2       = (float*)p;  p += (size_t)N * sizeof(float);
    int*   emax_ord = (int*)p;    p += (size_t)N * sizeof(int);
    float* denom    = (float*)p;

    gat_init<<<1024, 256, 0, stream>>>(out, emax_ord, denom, N);

    const int nwaves = (N + 15) / 16;
    gat_gemm<<<(nwaves + 7) / 8, 256, 0, stream>>>(inputs, fc_w, ft, N);

    gat_scores<<<(N + 255) / 256, 256, 0, stream>>>(ft, attn_l, attn_r, a1, a2, N);

    gat_edge_max<<<(E + 255) / 256, 256, 0, stream>>>(src, dst, a1, a2, emax_ord, E);
    gat_edge_denom<<<(E + 255) / 256, 256, 0, stream>>>(src, dst, a1, a2, emax_ord,
                                                        denom, E);

    gat_aggregate<<<(E + 7) / 8, 256, 0, stream>>>(src, dst, a1, a2, emax_ord, denom,
                                                   ft, out, E);
}